// PointSegLoss_44822278701826
// MI455X (gfx1250) — compile-verified
//
#include <hip/hip_runtime.h>
#include <stdint.h>

// PointSegLoss for MI455X (gfx1250, wave32).
// B=64, J=17, H=W=512.  Output: scalar f32 = pos_loss + neg_loss.
//
// Negative sampling = the num_neg smallest threefry-uniform values among
// eligible (outside-bbox) pixels, ties broken by pixel index (stable argsort).
// Exact via 2-level histogram (12+11 bits of the 23-bit key) + index-sorted
// tie list.  All cross-thread combining is integer atomics or fixed-order f32
// trees -> deterministic.  Final 32-lane sums use V_WMMA_F32_16X16X4_F32.
//
// Roofline choice: threefry (~75 int ops/px) x 3 passes is VALU-bound, while
// HBM is 23.3 TB/s -> if workspace is large enough we cache the 23-bit keys
// (64 MB) in pass 1 and the later passes stream 4 B/px instead of recomputing
// (USE_CACHE template; runtime-selected from ws_size, deterministic).

#define NB 64
#define NJ 17
#define HWPX 262144   // 512*512
#define WLOG 9        // W = 512
#define WMASK 511

// ---------------- workspace layout (bytes) ----------------
constexpr size_t OFF_HIST1   = 0;                                  // u32[64][4096]
constexpr size_t OFF_HIST2   = OFF_HIST1 + (size_t)NB * 4096 * 4;  // u32[64][2048]
constexpr size_t OFF_NUMPOS  = OFF_HIST2 + (size_t)NB * 2048 * 4;  // u32[64]
constexpr size_t OFF_NUMELIG = OFF_NUMPOS + NB * 4;
constexpr size_t OFF_EQCOUNT = OFF_NUMELIG + NB * 4;
constexpr size_t OFF_TOTPOS  = OFF_EQCOUNT + NB * 4;
constexpr size_t OFF_TOTNEG  = OFF_TOTPOS + 4;
constexpr size_t ZERO_END    = OFF_TOTNEG + 4;                     // zeroed each call
constexpr size_t OFF_BBOX    = (ZERO_END + 255) & ~(size_t)255;    // i32[64][4]
constexpr size_t OFF_NUMNEG  = OFF_BBOX + NB * 4 * 4;
constexpr size_t OFF_COARSE  = OFF_NUMNEG + NB * 4;
constexpr size_t OFF_QUOTA1  = OFF_COARSE + NB * 4;
constexpr size_t OFF_TKEY    = OFF_QUOTA1 + NB * 4;
constexpr size_t OFF_QUOTAEQ = OFF_TKEY + NB * 4;
constexpr size_t OFF_NEEDTIE = OFF_QUOTAEQ + NB * 4;
constexpr size_t OFF_CUTIDX  = OFF_NEEDTIE + NB * 4;
constexpr size_t OFF_EQLIST  = OFF_CUTIDX + NB * 4;                // u32[64][256]
constexpr size_t OFF_PART    = OFF_EQLIST + (size_t)NB * 256 * 4;  // f32[4096][2]
constexpr size_t WS_SMALL    = OFF_PART + (size_t)4096 * 2 * 4;    // ~1.7 MB
constexpr size_t OFF_KEYS    = (WS_SMALL + 255) & ~(size_t)255;    // u32[64][HW]
constexpr size_t WS_BIG      = OFF_KEYS + (size_t)NB * HWPX * 4;   // +64 MB

// ---------------- threefry2x32 (JAX, key(42) -> k0=0,k1=42) ----------------
__device__ __forceinline__ uint32_t rotl32(uint32_t x, uint32_t r) {
  return (x << r) | (x >> (32u - r));
}
// Partitionable counter scheme: element t uses counter (hi=0, lo=t),
// 32-bit bits = out0 ^ out1.  (Caveat: matches jax_threefry_partitionable.)
__device__ __forceinline__ uint32_t threefry_bits(uint32_t t) {
  const uint32_t k0 = 0u, k1 = 42u;
  const uint32_t k2 = 0x1BD11BDAu ^ k0 ^ k1;
  uint32_t x0 = k0;
  uint32_t x1 = t + k1;
#define TFR(r) { x0 += x1; x1 = rotl32(x1, r); x1 ^= x0; }
  TFR(13u) TFR(15u) TFR(26u) TFR(6u)
  x0 += k1; x1 += k2 + 1u;
  TFR(17u) TFR(29u) TFR(16u) TFR(24u)
  x0 += k2; x1 += k0 + 2u;
  TFR(13u) TFR(15u) TFR(26u) TFR(6u)
  x0 += k0; x1 += k1 + 3u;
  TFR(17u) TFR(29u) TFR(16u) TFR(24u)
  x0 += k1; x1 += k2 + 4u;
  TFR(13u) TFR(15u) TFR(26u) TFR(6u)
  x0 += k2; x1 += k0 + 5u;
#undef TFR
  return x0 ^ x1;
}
// uniform r = bitcast(0x3f800000 | (bits>>9)) - 1  -> ordering == key23 = bits>>9

__device__ __forceinline__ bool eligible_px(int idx, int ymn, int ymx, int xmn, int xmx) {
  int y = idx >> WLOG, x = idx & WMASK;
  bool inside = (y >= ymn) && (y < ymx) && (x >= xmn) && (x < xmx);
  return !inside;
}

__device__ __forceinline__ float softplusf(float x) {
  return fmaxf(x, 0.0f) + log1pf(expf(-fabsf(x)));
}

// ---------------- kernels ----------------
__global__ void k_zero(unsigned* __restrict__ ws, int nwords) {
  int i = blockIdx.x * blockDim.x + threadIdx.x;
  if (i < nwords) ws[i] = 0u;
}

__global__ void k_bbox(const float* __restrict__ skls, int* __restrict__ bbox) {
  int b = threadIdx.x;
  if (b >= NB) return;
  float xmn = 1e30f, xmx = -1e30f, ymn = 1e30f, ymx = -1e30f;
  for (int j = 0; j < NJ; ++j) {
    float x = skls[(b * NJ + j) * 2 + 0];
    float y = skls[(b * NJ + j) * 2 + 1];
    xmn = fminf(xmn, x); xmx = fmaxf(xmx, x);
    ymn = fminf(ymn, y); ymx = fmaxf(ymx, y);
  }
  bbox[b * 4 + 0] = (int)ymn;  // trunc toward zero (positive values)
  bbox[b * 4 + 1] = (int)ymx;
  bbox[b * 4 + 2] = (int)xmn;
  bbox[b * 4 + 3] = (int)xmx;
}

// Pass 1: num_pos, eligible count, coarse (top-12-bit) histogram; optionally
// cache the 23-bit keys for later passes (b128 stores).
template <bool CACHE>
__global__ __launch_bounds__(256) void k_coarse(const int* __restrict__ gt,
                                                const int* __restrict__ bbox,
                                                unsigned* __restrict__ hist1,
                                                unsigned* __restrict__ numPos,
                                                unsigned* __restrict__ numElig,
                                                unsigned* __restrict__ keys) {
  __shared__ unsigned h[4096];
  const int tid = threadIdx.x;
  const int b = blockIdx.y;
  for (int i = tid; i < 4096; i += 256) h[i] = 0u;
  __syncthreads();
  const int ymn = bbox[b * 4 + 0], ymx = bbox[b * 4 + 1];
  const int xmn = bbox[b * 4 + 2], xmx = bbox[b * 4 + 3];
  const int4* __restrict__ gt4 = (const int4*)(gt + (size_t)b * HWPX);
  uint4* __restrict__ key4 = CACHE ? (uint4*)(keys + (size_t)b * HWPX) : nullptr;
  unsigned pos = 0u, el = 0u;
  const int gbase = blockIdx.x * 1024;  // 1024 groups of 4 px per block
  for (int k = 0; k < 4; ++k) {
    int g4 = gbase + k * 256 + tid;
    int idx = g4 * 4;
    __builtin_prefetch(&gt4[g4 + 1024], 0, 0);  // global_prefetch_b8
    int4 g = gt4[g4];
    pos += (g.x > 0) + (g.y > 0) + (g.z > 0) + (g.w > 0);
    unsigned t0 = (unsigned)(b * HWPX + idx);
    if constexpr (CACHE) {
      // Compute all 4 unconditionally (wave pays for any active lane anyway)
      uint4 kk;
      kk.x = threefry_bits(t0 + 0u) >> 9;
      kk.y = threefry_bits(t0 + 1u) >> 9;
      kk.z = threefry_bits(t0 + 2u) >> 9;
      kk.w = threefry_bits(t0 + 3u) >> 9;
      key4[g4] = kk;  // global_store_b128
      unsigned ks[4] = {kk.x, kk.y, kk.z, kk.w};
#pragma unroll
      for (int j = 0; j < 4; ++j) {
        if (eligible_px(idx + j, ymn, ymx, xmn, xmx)) {
          ++el;
          atomicAdd(&h[ks[j] >> 11], 1u);  // top 12 of 23-bit key
        }
      }
    } else {
#pragma unroll
      for (int j = 0; j < 4; ++j) {
        if (eligible_px(idx + j, ymn, ymx, xmn, xmx)) {
          ++el;
          unsigned key = threefry_bits(t0 + (unsigned)j) >> 9;
          atomicAdd(&h[key >> 11], 1u);
        }
      }
    }
  }
  atomicAdd(&numPos[b], pos);
  atomicAdd(&numElig[b], el);
  __syncthreads();
  for (int i = tid; i < 4096; i += 256) {
    unsigned c = h[i];
    if (c) atomicAdd(&hist1[(size_t)b * 4096 + i], c);
  }
}

__global__ void k_scan_coarse(const unsigned* __restrict__ hist1,
                              const unsigned* __restrict__ numPos,
                              const unsigned* __restrict__ numElig,
                              unsigned* __restrict__ numNeg,
                              int* __restrict__ coarseBin,
                              unsigned* __restrict__ quota1,
                              unsigned* __restrict__ totPos,
                              unsigned* __restrict__ totNeg) {
  int b = threadIdx.x;
  if (b >= NB) return;
  unsigned np = numPos[b], ne = numElig[b];
  unsigned nn = np < ne ? np : ne;
  numNeg[b] = nn;
  unsigned cum = 0u, q1 = 0u;
  int cb = 0;
  if (nn > 0u) {
    for (int i = 0; i < 4096; ++i) {
      unsigned c = hist1[(size_t)b * 4096 + i];
      if (cum + c >= nn) { cb = i; q1 = nn - cum; break; }
      cum += c;
    }
  }
  coarseBin[b] = cb;
  quota1[b] = q1;
  atomicAdd(totPos, np);
  atomicAdd(totNeg, nn);
}

// Pass 2: fine (low-11-bit) histogram within the coarse threshold bin.
template <bool CACHE>
__global__ __launch_bounds__(256) void k_fine(const int* __restrict__ bbox,
                                              const int* __restrict__ coarseBin,
                                              const unsigned* __restrict__ numNeg,
                                              unsigned* __restrict__ hist2,
                                              const unsigned* __restrict__ keys) {
  const int b = blockIdx.y;
  if (numNeg[b] == 0u) return;  // uniform per block
  __shared__ unsigned h[2048];
  const int tid = threadIdx.x;
  for (int i = tid; i < 2048; i += 256) h[i] = 0u;
  __syncthreads();
  const int ymn = bbox[b * 4 + 0], ymx = bbox[b * 4 + 1];
  const int xmn = bbox[b * 4 + 2], xmx = bbox[b * 4 + 3];
  const unsigned cb = (unsigned)coarseBin[b];
  const uint4* __restrict__ key4 = CACHE ? (const uint4*)(keys + (size_t)b * HWPX) : nullptr;
  const int gbase = blockIdx.x * 1024;
  for (int k = 0; k < 4; ++k) {
    int g4 = gbase + k * 256 + tid;
    int idx = g4 * 4;
    unsigned ks[4];
    if constexpr (CACHE) {
      uint4 kk = key4[g4];  // 4 B/px stream instead of threefry recompute
      ks[0] = kk.x; ks[1] = kk.y; ks[2] = kk.z; ks[3] = kk.w;
    }
#pragma unroll
    for (int j = 0; j < 4; ++j) {
      if (eligible_px(idx + j, ymn, ymx, xmn, xmx)) {
        unsigned key;
        if constexpr (CACHE) key = ks[j];
        else key = threefry_bits((unsigned)(b * HWPX + idx + j)) >> 9;
        if ((key >> 11) == cb) atomicAdd(&h[key & 2047u], 1u);
      }
    }
  }
  __syncthreads();
  for (int i = tid; i < 2048; i += 256) {
    unsigned c = h[i];
    if (c) atomicAdd(&hist2[(size_t)b * 2048 + i], c);
  }
}

__global__ void k_scan_fine(const unsigned* __restrict__ hist2,
                            const unsigned* __restrict__ numNeg,
                            const int* __restrict__ coarseBin,
                            const unsigned* __restrict__ quota1,
                            unsigned* __restrict__ Tkey,
                            unsigned* __restrict__ quotaEq,
                            int* __restrict__ needTie,
                            unsigned* __restrict__ cutIdx) {
  int b = threadIdx.x;
  if (b >= NB) return;
  unsigned nn = numNeg[b];
  if (nn == 0u) { Tkey[b] = 0u; quotaEq[b] = 0u; needTie[b] = 0; cutIdx[b] = 0u; return; }
  unsigned q1 = quota1[b];
  unsigned cum = 0u, qe = 0u, ce = 0u;
  int fb = 0;
  for (int f = 0; f < 2048; ++f) {
    unsigned c = hist2[(size_t)b * 2048 + f];
    if (cum + c >= q1) { fb = f; qe = q1 - cum; ce = c; break; }
    cum += c;
  }
  Tkey[b] = ((unsigned)coarseBin[b] << 11) | (unsigned)fb;  // exact 23-bit threshold
  quotaEq[b] = qe;
  int nt = (qe < ce) ? 1 : 0;  // need index tie-break among equal keys
  needTie[b] = nt;
  if (!nt) cutIdx[b] = (unsigned)HWPX;  // select all equals
}

// Rare path (block-uniform early out): collect indices of key==T pixels.
__global__ __launch_bounds__(256) void k_tielist(const int* __restrict__ bbox,
                                                 const int* __restrict__ needTie,
                                                 const unsigned* __restrict__ Tkey,
                                                 unsigned* __restrict__ eqCount,
                                                 unsigned* __restrict__ eqList) {
  const int b = blockIdx.y;
  if (!needTie[b]) return;
  const int tid = threadIdx.x;
  const int ymn = bbox[b * 4 + 0], ymx = bbox[b * 4 + 1];
  const int xmn = bbox[b * 4 + 2], xmx = bbox[b * 4 + 3];
  const unsigned T = Tkey[b];
  const int base = blockIdx.x * 4096;
  for (int k = 0; k < 16; ++k) {
    int idx = base + k * 256 + tid;
    if (eligible_px(idx, ymn, ymx, xmn, xmx)) {
      unsigned key = threefry_bits((unsigned)(b * HWPX + idx)) >> 9;
      if (key == T) {
        unsigned p = atomicAdd(&eqCount[b], 1u);
        if (p < 256u) eqList[(size_t)b * 256 + p] = (unsigned)idx;
      }
    }
  }
}

__global__ void k_cut(const int* __restrict__ needTie,
                      const unsigned* __restrict__ eqCount,
                      unsigned* __restrict__ eqList,
                      const unsigned* __restrict__ quotaEq,
                      unsigned* __restrict__ cutIdx) {
  int b = threadIdx.x;
  if (b >= NB) return;
  if (!needTie[b]) return;
  unsigned n = eqCount[b];
  if (n > 256u) n = 256u;
  unsigned* L = &eqList[(size_t)b * 256];
  for (int i = 1; i < (int)n; ++i) {  // insertion sort -> deterministic
    unsigned v = L[i];
    int j = i - 1;
    while (j >= 0 && L[j] > v) { L[j + 1] = L[j]; --j; }
    L[j + 1] = v;
  }
  unsigned q = quotaEq[b];
  cutIdx[b] = (q < n) ? L[q] : (unsigned)HWPX;  // first q equals by index
}

// Pass 3: BCE terms; per-block deterministic partial sums.
template <bool CACHE>
__global__ __launch_bounds__(256) void k_loss(const float* __restrict__ masks,
                                              const int* __restrict__ gt,
                                              const int* __restrict__ bbox,
                                              const unsigned* __restrict__ Tkey,
                                              const unsigned* __restrict__ cutIdx,
                                              float* __restrict__ partials,
                                              const unsigned* __restrict__ keys) {
  __shared__ float sp[256], sn[256];
  const int tid = threadIdx.x;
  const int b = blockIdx.y;
  const int ymn = bbox[b * 4 + 0], ymx = bbox[b * 4 + 1];
  const int xmn = bbox[b * 4 + 2], xmx = bbox[b * 4 + 3];
  const unsigned T = Tkey[b];
  const unsigned cut = cutIdx[b];
  const int4*   __restrict__ gt4 = (const int4*)(gt + (size_t)b * HWPX);
  const float4* __restrict__ mk4 = (const float4*)(masks + (size_t)b * HWPX);
  const uint4*  __restrict__ key4 = CACHE ? (const uint4*)(keys + (size_t)b * HWPX) : nullptr;
  float pa = 0.0f, na = 0.0f;
  const int gbase = blockIdx.x * 1024;
  for (int k = 0; k < 4; ++k) {
    int g4 = gbase + k * 256 + tid;
    int idx = g4 * 4;
    __builtin_prefetch(&mk4[g4 + 1024], 0, 0);
    int4 g = gt4[g4];
    float4 m = mk4[g4];
    float ms[4] = {m.x, m.y, m.z, m.w};
    int gs[4] = {g.x, g.y, g.z, g.w};
    unsigned ks[4];
    if constexpr (CACHE) {
      uint4 kk = key4[g4];
      ks[0] = kk.x; ks[1] = kk.y; ks[2] = kk.z; ks[3] = kk.w;
    }
#pragma unroll
    for (int j = 0; j < 4; ++j) {
      if (gs[j] > 0) pa += softplusf(-ms[j]);
      if (eligible_px(idx + j, ymn, ymx, xmn, xmx)) {
        unsigned key;
        if constexpr (CACHE) key = ks[j];
        else key = threefry_bits((unsigned)(b * HWPX + idx + j)) >> 9;
        if (key < T || (key == T && (unsigned)(idx + j) < cut)) na += softplusf(ms[j]);
      }
    }
  }
  sp[tid] = pa; sn[tid] = na;
  __syncthreads();
  for (int s = 128; s > 0; s >>= 1) {  // fixed-order tree -> deterministic
    if (tid < s) { sp[tid] += sp[tid + s]; sn[tid] += sn[tid + s]; }
    __syncthreads();
  }
  if (tid == 0) {
    int bid = blockIdx.y * gridDim.x + blockIdx.x;
    partials[(size_t)bid * 2 + 0] = sp[0];
    partials[(size_t)bid * 2 + 1] = sn[0];
  }
}

// ---------------- WMMA 32-lane f32 reduction ----------------
typedef float v2f __attribute__((ext_vector_type(2)));
typedef float v8f __attribute__((ext_vector_type(8)));

// total = sum over all 32 lanes of v, via V_WMMA_F32_16X16X4_F32.
// Step 1: D = A(v,0) x ones; Step 2: acc += ones x B(D regs) chained over all
// 8 D regs -> sum of all A entries = sum of v.  Ones-operands make the result
// independent of the exact lane striping.
__device__ __forceinline__ float wave_sum32_wmma(float v) {
  v2f a;   a[0] = v;    a[1] = 0.0f;
  v2f one; one[0] = 1.0f; one[1] = 1.0f;
  v8f z = {0.f, 0.f, 0.f, 0.f, 0.f, 0.f, 0.f, 0.f};
  v8f d = __builtin_amdgcn_wmma_f32_16x16x4_f32(false, a, false, one, (short)0, z,
                                                false, false);
  v8f acc = {0.f, 0.f, 0.f, 0.f, 0.f, 0.f, 0.f, 0.f};
  v2f bb;
  bb[0] = d[0]; bb[1] = d[1];
  acc = __builtin_amdgcn_wmma_f32_16x16x4_f32(false, one, false, bb, (short)0, acc,
                                              false, false);
  bb[0] = d[2]; bb[1] = d[3];
  acc = __builtin_amdgcn_wmma_f32_16x16x4_f32(false, one, false, bb, (short)0, acc,
                                              false, false);
  bb[0] = d[4]; bb[1] = d[5];
  acc = __builtin_amdgcn_wmma_f32_16x16x4_f32(false, one, false, bb, (short)0, acc,
                                              false, false);
  bb[0] = d[6]; bb[1] = d[7];
  acc = __builtin_amdgcn_wmma_f32_16x16x4_f32(false, one, false, bb, (short)0, acc,
                                              false, false);
  return acc[0];
}

__global__ __launch_bounds__(1024) void k_reduce(const float* __restrict__ partials,
                                                 const unsigned* __restrict__ totPos,
                                                 const unsigned* __restrict__ totNeg,
                                                 float* __restrict__ out) {
  __shared__ float sp[1024], sn[1024];
  const int tid = threadIdx.x;
  float p = 0.0f, n = 0.0f;
  for (int i = tid; i < 4096; i += 1024) {  // fixed order per thread
    p += partials[(size_t)i * 2 + 0];
    n += partials[(size_t)i * 2 + 1];
  }
  sp[tid] = p; sn[tid] = n;
  __syncthreads();
  for (int s = 512; s >= 32; s >>= 1) {
    if (tid < s) { sp[tid] += sp[tid + s]; sn[tid] += sn[tid + s]; }
    __syncthreads();
  }
  // Every thread participates (EXEC all-ones required for WMMA); all waves
  // redundantly reduce sp[0..31]/sn[0..31].
  float vp = sp[tid & 31];
  float vn = sn[tid & 31];
  float tp = wave_sum32_wmma(vp);
  float tn = wave_sum32_wmma(vn);
  if (tid == 0) {
    float dp = (float)(*totPos);
    float dn = (float)(*totNeg);
    out[0] = tp / dp + tn / dn;
  }
}

// ---------------- launch ----------------
extern "C" void kernel_launch(void* const* d_in, const int* in_sizes, int n_in,
                              void* d_out, int out_size, void* d_ws, size_t ws_size,
                              hipStream_t stream) {
  (void)in_sizes; (void)n_in; (void)out_size;
  const float* masks = (const float*)d_in[0];
  const float* skls  = (const float*)d_in[1];
  const int*   gt    = (const int*)d_in[2];
  float* out = (float*)d_out;
  char* w = (char*)d_ws;

  unsigned* hist1   = (unsigned*)(w + OFF_HIST1);
  unsigned* hist2   = (unsigned*)(w + OFF_HIST2);
  unsigned* numPos  = (unsigned*)(w + OFF_NUMPOS);
  unsigned* numElig = (unsigned*)(w + OFF_NUMELIG);
  unsigned* eqCount = (unsigned*)(w + OFF_EQCOUNT);
  unsigned* totPos  = (unsigned*)(w + OFF_TOTPOS);
  unsigned* totNeg  = (unsigned*)(w + OFF_TOTNEG);
  int*      bbox    = (int*)(w + OFF_BBOX);
  unsigned* numNeg  = (unsigned*)(w + OFF_NUMNEG);
  int*      coarse  = (int*)(w + OFF_COARSE);
  unsigned* quota1  = (unsigned*)(w + OFF_QUOTA1);
  unsigned* Tkey    = (unsigned*)(w + OFF_TKEY);
  unsigned* quotaEq = (unsigned*)(w + OFF_QUOTAEQ);
  int*      needTie = (int*)(w + OFF_NEEDTIE);
  unsigned* cutIdx  = (unsigned*)(w + OFF_CUTIDX);
  unsigned* eqList  = (unsigned*)(w + OFF_EQLIST);
  float*    part    = (float*)(w + OFF_PART);
  unsigned* keys    = (unsigned*)(w + OFF_KEYS);

  const bool useCache = (ws_size >= WS_BIG);  // constant per harness -> deterministic

  const int zwords = (int)(ZERO_END / 4);
  k_zero<<<(zwords + 255) / 256, 256, 0, stream>>>((unsigned*)w, zwords);
  k_bbox<<<1, 64, 0, stream>>>(skls, bbox);

  dim3 grid(64, NB);  // 64 blocks * 4096 px = 262144 px per image
  if (useCache) {
    k_coarse<true><<<grid, 256, 0, stream>>>(gt, bbox, hist1, numPos, numElig, keys);
  } else {
    k_coarse<false><<<grid, 256, 0, stream>>>(gt, bbox, hist1, numPos, numElig, keys);
  }
  k_scan_coarse<<<1, 64, 0, stream>>>(hist1, numPos, numElig, numNeg, coarse,
                                      quota1, totPos, totNeg);
  if (useCache) {
    k_fine<true><<<grid, 256, 0, stream>>>(bbox, coarse, numNeg, hist2, keys);
  } else {
    k_fine<false><<<grid, 256, 0, stream>>>(bbox, coarse, numNeg, hist2, keys);
  }
  k_scan_fine<<<1, 64, 0, stream>>>(hist2, numNeg, coarse, quota1, Tkey, quotaEq,
                                    needTie, cutIdx);
  k_tielist<<<grid, 256, 0, stream>>>(bbox, needTie, Tkey, eqCount, eqList);
  k_cut<<<1, 64, 0, stream>>>(needTie, eqCount, eqList, quotaEq, cutIdx);
  if (useCache) {
    k_loss<true><<<grid, 256, 0, stream>>>(masks, gt, bbox, Tkey, cutIdx, part, keys);
  } else {
    k_loss<false><<<grid, 256, 0, stream>>>(masks, gt, bbox, Tkey, cutIdx, part, keys);
  }
  k_reduce<<<1, 1024, 0, stream>>>(part, totPos, totNeg, out);
}